// DeepAR_17557826306419
// MI455X (gfx1250) — compile-verified
//
#include <hip/hip_runtime.h>
#include <math.h>

// DeepAR fused recurrence for MI455X (gfx1250, wave32).
// One workgroup / one wave32; whole model state + weights live in LDS.
// Attention matmul (32x32x32, f32) runs on V_WMMA_F32_16X16X4_F32.

#define S_LEN 512
#define HORIZ 64
#define T_LEN (S_LEN + HORIZ)   // 576
#define F_IN  4
#define E_EMB 32
#define HID   32
#define NG    (4 * HID)          // 128 gate rows
#define XDIM  (F_IN + E_EMB)     // 36 LSTM input width

typedef __attribute__((ext_vector_type(2))) float v2f;
typedef __attribute__((ext_vector_type(8))) float v8f;

__device__ __forceinline__ float sigmoidf_fast(float x) {
  return 1.0f / (1.0f + __expf(-x));
}

__launch_bounds__(32, 1)
__global__ void deepar_fused(const float* __restrict__ X,   const float* __restrict__ y,
                             const float* __restrict__ Xf,  const float* __restrict__ eps,
                             const float* __restrict__ W_ye, const float* __restrict__ b_ye,
                             const float* __restrict__ W_ih, const float* __restrict__ W_hh,
                             const float* __restrict__ b_ih, const float* __restrict__ b_hh,
                             const float* __restrict__ W_ef, const float* __restrict__ b_ef,
                             const float* __restrict__ W_av, const float* __restrict__ b_av,
                             const float* __restrict__ W_out, const float* __restrict__ b_out,
                             const float* __restrict__ W_mu, const float* __restrict__ b_mu,
                             const float* __restrict__ W_sig, const float* __restrict__ b_sig,
                             float* __restrict__ out)
{
  // ---- LDS working set (~61 KB, fits comfortably in the 320 KB WGP pool) ----
  __shared__ float sWih[NG * XDIM];      // 128x36 LSTM input weights
  __shared__ float sWhh[NG * HID];       // 128x32 LSTM recurrent weights
  __shared__ float sBg[NG];              // b_ih + b_hh fused
  __shared__ float sWye[E_EMB], sBye[E_EMB];
  __shared__ float sWef[32], sBef[32], sWav[32], sWoutW[64], sWmu[32], sWsig[32];
  __shared__ float sH[HID];              // hidden state (broadcast copy)
  __shared__ float sX[XDIM];             // [x_t(4) | yemb(32)]
  __shared__ float sHs[32 * 32];         // hs_e  (row m = hidden unit, col j = embed)
  __shared__ float sAw[32 * 32];         // attention weights (strict lower-tri, normalized)
  __shared__ float sWo[32 * 32];         // wout = aw @ hs_e (WMMA result)
  __shared__ float sAv[32];              // exp attention scores
  __shared__ float sXall[T_LEN * F_IN];  // X ++ Xf staged once
  __shared__ float sYE[S_LEN];           // observed y
  __shared__ float sEps[T_LEN];          // Gaussian noise

  const int lane = (int)threadIdx.x;     // wave32: lane == thread

  // ---- One-time stage of all weights & streams into LDS ----
  for (int i = lane; i < NG * XDIM; i += 32) sWih[i] = W_ih[i];
  for (int i = lane; i < NG * HID;  i += 32) sWhh[i] = W_hh[i];
  for (int i = lane; i < NG;        i += 32) sBg[i]  = b_ih[i] + b_hh[i];
  for (int i = lane; i < S_LEN * F_IN; i += 32) sXall[i] = X[i];
  for (int i = lane; i < HORIZ * F_IN; i += 32) sXall[S_LEN * F_IN + i] = Xf[i];
  for (int i = lane; i < S_LEN; i += 32) sYE[i]  = y[i];
  for (int i = lane; i < T_LEN; i += 32) sEps[i] = eps[i];
  sWye[lane] = W_ye[lane];  sBye[lane] = b_ye[lane];
  sWef[lane] = W_ef[lane];  sBef[lane] = b_ef[lane];
  sWav[lane] = W_av[lane];
  sWoutW[lane] = W_out[lane];  sWoutW[lane + 32] = W_out[lane + 32];
  sWmu[lane] = W_mu[lane];  sWsig[lane] = W_sig[lane];
  sH[lane] = 0.0f;
  const float b_av0 = b_av[0], b_out0 = b_out[0], b_mu0 = b_mu[0], b_sig0 = b_sig[0];

  float c_reg  = 0.0f;   // cell state for hidden unit `lane`
  float y_prev = 0.0f;   // previous Gaussian sample (uniform across lanes)
  __syncthreads();

  const int halfsel = lane >> 4;   // 0: lanes 0-15, 1: lanes 16-31
  const int l16     = lane & 15;

  for (int t = 0; t < T_LEN; ++t) {
    // ---- input embed: yin -> yemb (lane = embed index) ----
    float yin = (t < S_LEN) ? sYE[t] : y_prev;
    if (lane < F_IN) sX[lane] = sXall[t * F_IN + lane];
    sX[F_IN + lane] = fmaf(yin, sWye[lane], sBye[lane]);
    __syncthreads();

    // ---- LSTM cell: lane computes gate rows {lane, +32, +64, +96} = i,f,g,o ----
    float gi = sBg[lane], gf = sBg[lane + 32], gg = sBg[lane + 64], go = sBg[lane + 96];
    #pragma unroll
    for (int k = 0; k < XDIM; ++k) {
      float xv = sX[k];
      gi = fmaf(xv, sWih[(lane      ) * XDIM + k], gi);
      gf = fmaf(xv, sWih[(lane + 32 ) * XDIM + k], gf);
      gg = fmaf(xv, sWih[(lane + 64 ) * XDIM + k], gg);
      go = fmaf(xv, sWih[(lane + 96 ) * XDIM + k], go);
    }
    #pragma unroll
    for (int k = 0; k < HID; ++k) {
      float hv = sH[k];
      gi = fmaf(hv, sWhh[(lane      ) * HID + k], gi);
      gf = fmaf(hv, sWhh[(lane + 32 ) * HID + k], gf);
      gg = fmaf(hv, sWhh[(lane + 64 ) * HID + k], gg);
      go = fmaf(hv, sWhh[(lane + 96 ) * HID + k], go);
    }
    float ig = sigmoidf_fast(gi), fg = sigmoidf_fast(gf), og = sigmoidf_fast(go);
    c_reg = fmaf(fg, c_reg, ig * tanhf(gg));
    float h_new = og * tanhf(c_reg);
    __syncthreads();              // all lanes done reading old sH
    sH[lane] = h_new;
    __syncthreads();

    // ---- hs_e[m][j] = h[m]*W_ef[j] + b_ef[j]; lane owns column j = lane ----
    {
      float wj = sWef[lane], bj = sBef[lane];
      #pragma unroll
      for (int m = 0; m < HID; ++m) sHs[m * 32 + lane] = fmaf(sH[m], wj, bj);
    }
    __syncthreads();

    // ---- attention scores av[m] = exp(hs_e[m,:] . W_av + b_av) ----
    float avm = b_av0;
    #pragma unroll
    for (int j = 0; j < 32; ++j) avm = fmaf(sHs[lane * 32 + j], sWav[j], avm);
    sAv[lane] = __expf(avm);
    __syncthreads();

    // ---- strict-lower-triangular normalized weights: aw[m][j] ----
    float acc = 1e-9f;
    for (int j = 0; j < lane; ++j) acc += sAv[j];
    float inv = 1.0f / acc;
    #pragma unroll
    for (int j = 0; j < 32; ++j) sAw[lane * 32 + j] = (j < lane) ? sAv[j] * inv : 0.0f;
    __syncthreads();

    // ---- wout = aw(32x32) @ hs_e(32x32) on V_WMMA_F32_16X16X4_F32 ----
    // 4 output tiles x 8 k-chunks = 32 f32 WMMAs per step, full fp32 precision.
    // A(16x4):  a.x/a.y hold K = k0+2*halfsel, +1  for row M = tm*16 + l16
    // B(4x16):  b.x/b.y hold rows K = k0+2*halfsel, +1 at col N = tn*16 + l16
    // D(16x16): element v of the v8f is row tm*16 + v + 8*halfsel, col tn*16 + l16
    #pragma unroll
    for (int tm = 0; tm < 2; ++tm) {
      #pragma unroll
      for (int tn = 0; tn < 2; ++tn) {
        v8f d = {};
        #pragma unroll
        for (int k0 = 0; k0 < 32; k0 += 4) {
          int ka  = k0 + 2 * halfsel;
          int row = tm * 16 + l16;
          int col = tn * 16 + l16;
          v2f a; a.x = sAw[row * 32 + ka];      a.y = sAw[row * 32 + ka + 1];
          v2f b; b.x = sHs[ka * 32 + col];      b.y = sHs[(ka + 1) * 32 + col];
          d = __builtin_amdgcn_wmma_f32_16x16x4_f32(
                false, a, false, b, (short)0, d, false, false);
        }
        #pragma unroll
        for (int v = 0; v < 8; ++v) {
          int row = tm * 16 + v + 8 * halfsel;
          int col = tn * 16 + l16;
          sWo[row * 32 + col] = d[v];
        }
      }
    }
    __syncthreads();

    // ---- out[m] = tanh([hs_e | wout] . W_out + b_out) ----
    float acco = b_out0;
    #pragma unroll
    for (int k = 0; k < 32; ++k) acco = fmaf(sHs[lane * 32 + k], sWoutW[k], acco);
    #pragma unroll
    for (int k = 0; k < 32; ++k) acco = fmaf(sWo[lane * 32 + k], sWoutW[32 + k], acco);
    float om = tanhf(acco);

    // ---- mu / sigma heads: wave32 xor-shuffle tree reduction ----
    float pm = om * sWmu[lane];
    float ps = om * sWsig[lane];
    #pragma unroll
    for (int off = 16; off > 0; off >>= 1) {
      pm += __shfl_xor(pm, off, 32);
      ps += __shfl_xor(ps, off, 32);
    }
    float mu = pm + b_mu0;
    float sg = log1pf(__expf(ps + b_sig0)) + 1e-6f;
    float ynew = fmaf(sg, sEps[t], mu);   // detached Gaussian sample
    y_prev = ynew;                         // uniform across lanes (full reduction)

    if (lane == 0) {
      out[HORIZ + t]         = mu;   // mu block
      out[HORIZ + T_LEN + t] = sg;   // sigma block
      if (t >= S_LEN - 1 && t < S_LEN - 1 + HORIZ)
        out[t - (S_LEN - 1)] = ynew; // ypred block
    }
    __syncthreads();   // protect LDS reuse before next iteration
  }
}

extern "C" void kernel_launch(void* const* d_in, const int* in_sizes, int n_in,
                              void* d_out, int out_size, void* d_ws, size_t ws_size,
                              hipStream_t stream) {
  (void)in_sizes; (void)n_in; (void)out_size; (void)d_ws; (void)ws_size;
  const float* X     = (const float*)d_in[0];
  const float* y     = (const float*)d_in[1];
  const float* Xf    = (const float*)d_in[2];
  const float* eps   = (const float*)d_in[3];
  const float* W_ye  = (const float*)d_in[4];
  const float* b_ye  = (const float*)d_in[5];
  const float* W_ih  = (const float*)d_in[6];
  const float* W_hh  = (const float*)d_in[7];
  const float* b_ih  = (const float*)d_in[8];
  const float* b_hh  = (const float*)d_in[9];
  const float* W_ef  = (const float*)d_in[10];
  const float* b_ef  = (const float*)d_in[11];
  const float* W_av  = (const float*)d_in[12];
  const float* b_av  = (const float*)d_in[13];
  const float* W_out = (const float*)d_in[14];
  const float* b_out = (const float*)d_in[15];
  const float* W_mu  = (const float*)d_in[16];
  const float* b_mu  = (const float*)d_in[17];
  const float* W_sig = (const float*)d_in[18];
  const float* b_sig = (const float*)d_in[19];

  // Single wave32 workgroup: the recurrence is one serial dependence chain;
  // everything lives in LDS on one WGP.
  deepar_fused<<<1, 32, 0, stream>>>(X, y, Xf, eps, W_ye, b_ye, W_ih, W_hh,
                                     b_ih, b_hh, W_ef, b_ef, W_av, b_av,
                                     W_out, b_out, W_mu, b_mu, W_sig, b_sig,
                                     (float*)d_out);
}